// PermutationMatrixLearningModule_18872086299191
// MI455X (gfx1250) — compile-verified
//
#include <hip/hip_runtime.h>
#include <math.h>

#define NN 64
#define APAD 65          // padded LDS row stride (floats): bank = (p+j)%64, conflict-free
#define NUM_PHASES 8
#define MAX_ITERS 2000
#define THETA 0.25f
#define EPS_FINAL 1e-3f

typedef unsigned int u32x4 __attribute__((ext_vector_type(4)));
typedef int          i32x8 __attribute__((ext_vector_type(8)));
typedef int          i32x4 __attribute__((ext_vector_type(4)));

// One workgroup (64 threads = 2 wave32) per batch element.
// Thread t plays the role of person t (bid step) and object t (resolve step).
__global__ __launch_bounds__(64) void auction_lap_kernel(
    const float* __restrict__ gamma_raw,   // [64,64]
    const float* __restrict__ u,           // [B,64,64]
    float* __restrict__ out)               // [B,64,64] one-hot permutation
{
    __shared__ float u_raw[NN * NN];                 // 16 KB raw staging (TDM target)
    __shared__ float a_sh[NN * APAD];                // padded logits tile
    __shared__ float price[NN];
    __shared__ unsigned long long slot[NN];          // per-object best bid (packed)
    __shared__ int owner_obj[NN];                    // object -> person
    __shared__ int assigned_obj[NN];                 // person -> object
    __shared__ float redmax[NN];

    const int t = threadIdx.x;                       // 0..63
    const int b = blockIdx.x;
    const float* ubase = u + (size_t)b * (NN * NN);

    // ---- Kick off the 16KB u-tile DMA (Tensor Data Mover), then overlap with
    //      the sigmoid(gamma) pass before waiting. ----
#if __has_builtin(__builtin_amdgcn_tensor_load_to_lds)
    {
        const unsigned lds_addr = (unsigned)(size_t)(&u_raw[0]);
        const unsigned long long ga = (unsigned long long)(size_t)ubase;
        // D# group 0: count=1 (valid, user mode), lds_addr, global_addr[56:0], type=2
        u32x4 g0;
        g0.x = 1u;
        g0.y = lds_addr;
        g0.z = (unsigned)(ga & 0xffffffffu);
        g0.w = (unsigned)((ga >> 32) & 0x01ffffffu) | (2u << 30);
        // D# group 1: mask=0, data_size=4B(2), dims/tile 64x64, stride0=64
        i32x8 g1;
        g1[0] = 0x00020000;        // [17:16] data_size = 2 -> 4 bytes
        g1[1] = NN << 16;          // tensor_dim0[15:0] in bits [63:48]
        g1[2] = NN << 16;          // tensor_dim0 hi=0; tensor_dim1[15:0] in [95:80]... bits [31:16]
        g1[3] = NN << 16;          // tensor_dim1 hi=0; tile_dim0=64 in bits [127:112]
        g1[4] = NN;                // tile_dim1=64; tile_dim2=0
        g1[5] = NN;                // tensor_dim0_stride[31:0] = 64
        g1[6] = 0;
        g1[7] = 0;
        i32x4 z4 = {0, 0, 0, 0};   // groups 2/3: unused (2-D tile)
        if (t < 32) {              // TDM ignores EXEC: gate by branch, wave 0 only
#if defined(__clang_major__) && (__clang_major__ >= 23)
            i32x8 z8 = {0, 0, 0, 0, 0, 0, 0, 0};
            __builtin_amdgcn_tensor_load_to_lds(g0, g1, z4, z4, z8, 0);
#else
            __builtin_amdgcn_tensor_load_to_lds(g0, g1, z4, z4, 0);
#endif
        }
    }
#else
    // Fallback: async global->LDS b128 copies (16 per thread)
    {
        unsigned lds_base = (unsigned)(size_t)(&u_raw[0]);
#pragma unroll
        for (int kk = 0; kk < 16; ++kk) {
            unsigned byteoff = (unsigned)t * 16u + (unsigned)kk * 1024u;
            unsigned ldsa = lds_base + byteoff;
            asm volatile(
                "global_load_async_to_lds_b128 %0, %1, %2 offset:0"
                :: "v"(ldsa), "v"(byteoff), "s"(ubase) : "memory");
        }
    }
#endif
    __builtin_prefetch(gamma_raw, 0, 0);             // global_prefetch_b8 (L2-resident tile)

    // ---- Overlapped pass 1: a_sh <- sigmoid(gamma) while the DMA runs ----
    for (int k = 0; k < NN; ++k) {
        float g = 1.0f / (1.0f + expf(-gamma_raw[k * NN + t]));  // coalesced
        a_sh[k * APAD + t] = g;
    }

#if __has_builtin(__builtin_amdgcn_tensor_load_to_lds)
    __builtin_amdgcn_s_wait_tensorcnt(0);            // no-op on wave 1 (TENSORcnt==0)
#else
    asm volatile("s_wait_asynccnt 0" ::: "memory");
#endif
    __syncthreads();                                 // DMA visible to both waves

    // ---- Pass 2: a += Gumbel(u) = -log(-log(u)); track max|a| ----
    float mloc = 0.0f;
    for (int k = 0; k < NN; ++k) {
        float uv  = u_raw[k * NN + t];               // bank = t, conflict-free
        float val = a_sh[k * APAD + t] - logf(-logf(uv));
        a_sh[k * APAD + t] = val;
        mloc = fmaxf(mloc, fabsf(val));
    }
    redmax[t] = mloc;
    price[t] = 0.0f;
    __syncthreads();
    for (int s = 32; s > 0; s >>= 1) {
        if (t < s) redmax[t] = fmaxf(redmax[t], redmax[t + s]);
        __syncthreads();
    }
    const float eps0 = fmaxf(0.5f * redmax[0], EPS_FINAL);

    // ---- epsilon-scaled Jacobi auction ----
    float epsv = eps0;
    for (int phase = 0; phase < NUM_PHASES; ++phase) {
        const float eps = fmaxf(epsv, EPS_FINAL);
        epsv *= THETA;

        owner_obj[t] = -1;
        assigned_obj[t] = -1;
        slot[t] = 0ull;
        __syncthreads();

        int unassigned = NN;
        int it = 0;
        while (unassigned > 0 && it < MAX_ITERS) {
            // -- bid step (thread = person) --
            if (assigned_obj[t] < 0) {
                const float* arow = &a_sh[t * APAD];
                float v1 = -3.0e38f, v2 = -3.0e38f;
                int j1 = 0;
                for (int j = 0; j < NN; ++j) {
                    float v = arow[j] - price[j];      // price[j] broadcast read
                    if (v > v1) { v2 = v1; v1 = v; j1 = j; }
                    else if (v > v2) { v2 = v; }
                }
                float bid = v1 - v2 + eps;             // > 0 => float bits order-preserving
                unsigned long long key =
                    ((unsigned long long)__float_as_uint(bid) << 32) |
                    (unsigned long long)(unsigned)(63 - t);   // ties -> lowest person wins
                atomicMax(&slot[j1], key);             // ds_max_u64 arbitration
            }
            __syncthreads();

            // -- resolve step (thread = object); also reset own slot for next round --
            unsigned long long k = slot[t];
            slot[t] = 0ull;                            // safe: t owns slot[t]; bids done
            if (k != 0ull) {
                float bid = __uint_as_float((unsigned)(k >> 32));
                int w = 63 - (int)(k & 0xffffffffu);
                price[t] += bid;
                int prev = owner_obj[t];
                owner_obj[t] = w;
                if (prev >= 0) assigned_obj[prev] = -1;  // prev owned exactly this object
                assigned_obj[w] = t;                     // w bid on exactly one object
            }
            __syncthreads();

            unassigned = __syncthreads_count(assigned_obj[t] < 0);
            ++it;
        }
    }
    __syncthreads();

    // ---- one-hot output, fully coalesced: lane t writes column t of each row ----
    float* orow = out + (size_t)b * (NN * NN);
    for (int r = 0; r < NN; ++r) {
        int col = assigned_obj[r];
        col = (col < 0) ? 0 : col;        // matches jnp.argmax(all-false) == 0
        orow[r * NN + t] = (col == t) ? 1.0f : 0.0f;
    }
}

extern "C" void kernel_launch(void* const* d_in, const int* in_sizes, int n_in,
                              void* d_out, int out_size, void* d_ws, size_t ws_size,
                              hipStream_t stream) {
    const float* gamma_raw = (const float*)d_in[0];   // [64,64]
    const float* u         = (const float*)d_in[1];   // [B,64,64]
    float* out             = (float*)d_out;
    (void)d_ws; (void)ws_size; (void)n_in; (void)out_size;

    const int B = in_sizes[1] / (NN * NN);
    auction_lap_kernel<<<B, NN, 0, stream>>>(gamma_raw, u, out);
}